// FMAPModelWarping_3556232921145
// MI455X (gfx1250) — compile-verified
//
#include <hip/hip_runtime.h>
#include <hip/hip_bf16.h>
#include <math.h>

typedef __attribute__((ext_vector_type(2))) float v2f;
typedef __attribute__((ext_vector_type(8))) float v8f;

#define HH 60
#define WW 60
#define PIX 3600           // 60*60
#define BD 4               // batch
#define DF 128             // FEAT
#define HID 64
#define NT 225             // 3600/16 tiles
#define MT 5               // m-subtiles per wave strip (225 = 45*5, no tail)
#define NSTRIP 45          // m strips

// ---- workspace layout (in floats) ----
#define MATS_OFF   ((size_t)0)                    // 16 * 12
#define XW_OFF     ((size_t)192)                  // 4*4*3*3600   = 172800
#define Y_OFF      (XW_OFF + (size_t)172800)      // 4*4*128*3600 = 7372800
#define F_OFF      (Y_OFF + (size_t)7372800)
#define HMAX_OFF   (F_OFF + (size_t)7372800)      // B*3600 (uint)
#define HSUM_OFF   (HMAX_OFF + (size_t)14400)
#define VMAX_OFF   (HSUM_OFF + (size_t)14400)     // (uint)
#define VSUM_OFF   (VMAX_OFF + (size_t)14400)
#define S_OFF      (VSUM_OFF + (size_t)14400)

// xor-lane swap in one ds_swizzle_b32 (group-of-32: and=0x1F, xor=mask)
#define SWZ(v, mask) __int_as_float(__builtin_amdgcn_ds_swizzle(__float_as_int(v), (((mask) << 10) | 0x1F)))

// monotonic float<->uint encoding so atomicMax(uint) == float max
__device__ __forceinline__ unsigned fenc(float f) {
    unsigned u = __float_as_uint(f);
    return (u & 0x80000000u) ? ~u : (u | 0x80000000u);
}
__device__ __forceinline__ float fdec(unsigned u) {
    return __uint_as_float((u & 0x80000000u) ? (u & 0x7FFFFFFFu) : ~u);
}

// ---------------- setup: affine matrices A (fwd-or-inv) and Bm ----------------
__global__ void make_mats(const float* __restrict__ noise, const int* __restrict__ swp,
                          float* __restrict__ mats) {
    int t = threadIdx.x;
    if (t >= 16) return;
    int w = t >> 2, b = t & 3;
    const float* nz = noise + (size_t)(w * 4 + b) * 9;
    // fwd = I + 0.05*noise with last row forced to [0,0,1]
    float a  = 1.f + 0.05f * nz[0], bb = 0.05f * nz[1], c = 0.05f * nz[2];
    float d  = 0.05f * nz[3],       e  = 1.f + 0.05f * nz[4], f = 0.05f * nz[5];
    float id = 1.f / (a * e - bb * d);
    float fwd6[6] = { a, bb, c, d, e, f };
    float inv6[6] = { e * id, -bb * id, (bb * f - c * e) * id,
                      -d * id,  a * id, (c * d - a * f) * id };
    bool s = (swp[w] == 1);
    float* A  = mats + (size_t)t * 12;
    float* Bm = A + 6;
    #pragma unroll
    for (int k = 0; k < 6; k++) { A[k] = s ? inv6[k] : fwd6[k]; Bm[k] = s ? fwd6[k] : inv6[k]; }
}

__global__ void zero_accum(float* __restrict__ p, int n) {
    int i = blockIdx.x * blockDim.x + threadIdx.x;
    if (i < n) p[i] = 0.0f;
}

// ---------------- front sample: xw = grid_sample(roll(x), grid(A)) ----------------
__global__ void sample_front(const float* __restrict__ ia, const float* __restrict__ ib,
                             const int* __restrict__ u_roll, const int* __restrict__ v_roll,
                             const float* __restrict__ mats, float* __restrict__ xw) {
    int idx = blockIdx.x * blockDim.x + threadIdx.x;
    if (idx >= 16 * PIX) return;
    int w   = idx / (4 * PIX);
    int rem = idx - w * 4 * PIX;
    int b   = rem / PIX;
    int p   = rem - b * PIX;
    int i = p / WW, j = p % WW;
    int wb = w * 4 + b;
    const float* A = mats + (size_t)wb * 12;
    int u = u_roll[wb], v = v_roll[wb];
    const float* img = (w & 1) ? ib : ia;

    float xs = (2.f * j + 1.f) / (float)WW - 1.f;
    float ys = (2.f * i + 1.f) / (float)HH - 1.f;
    float gx = A[0] * xs + A[1] * ys + A[2];
    float gy = A[3] * xs + A[4] * ys + A[5];
    float fx = (gx + 1.f) * (WW * 0.5f) - 0.5f;
    float fy = (gy + 1.f) * (HH * 0.5f) - 0.5f;
    float x0f = floorf(fx), y0f = floorf(fy);
    int x0 = (int)x0f, y0 = (int)y0f;
    float wx1 = fx - x0f, wx0 = 1.f - wx1;
    float wy1 = fy - y0f, wy0 = 1.f - wy1;

    for (int c = 0; c < 3; c++) {
        const float* ip = img + ((size_t)b * 3 + c) * PIX;
        auto samp = [&](int ix, int iy) -> float {
            if (ix < 0 || ix >= WW || iy < 0 || iy >= HH) return 0.f;
            int ri = iy + u; if (ri >= HH) ri -= HH;   // fused roll
            int rj = ix + v; if (rj >= WW) rj -= WW;
            return ip[ri * WW + rj];
        };
        float acc = samp(x0, y0)     * wx0 * wy0
                  + samp(x0+1, y0)   * wx1 * wy0
                  + samp(x0, y0+1)   * wx0 * wy1
                  + samp(x0+1, y0+1) * wx1 * wy1;
        xw[((size_t)wb * 3 + c) * PIX + p] = acc;
    }
}

// ---------------- fused conv3x3(3->64)+ReLU and conv1x1(64->128) ----------------
__global__ __launch_bounds__(64) void conv_fused(const float* __restrict__ xw,
                                                 const float* __restrict__ w1, const float* __restrict__ b1,
                                                 const float* __restrict__ w2, const float* __restrict__ b2,
                                                 float* __restrict__ y) {
    __shared__ float hid[HID];
    int blk = blockIdx.x;                 // (w*4+b)*3600 + p
    int wb = blk / PIX;
    int p  = blk - wb * PIX;
    int i = p / WW, j = p % WW;
    int o = threadIdx.x;
    const float* xbase = xw + (size_t)wb * 3 * PIX;

    float acc = b1[o];
    #pragma unroll
    for (int c = 0; c < 3; c++)
        #pragma unroll
        for (int ky = 0; ky < 3; ky++) {
            int ii = i + ky - 1;
            if (ii < 0 || ii >= HH) continue;
            #pragma unroll
            for (int kx = 0; kx < 3; kx++) {
                int jj = j + kx - 1;
                if (jj < 0 || jj >= WW) continue;
                acc += w1[((o * 3 + c) * 3 + ky) * 3 + kx] * xbase[c * PIX + ii * WW + jj];
            }
        }
    hid[o] = fmaxf(acc, 0.f);
    __syncthreads();

    for (int t = o; t < DF; t += HID) {
        float f = b2[t];
        const float* wr = w2 + (size_t)t * HID;
        #pragma unroll 8
        for (int c2 = 0; c2 < HID; c2++) f += wr[c2] * hid[c2];
        y[((size_t)wb * DF + t) * PIX + p] = f;
    }
}

// ---------------- back sample + un-roll into WMMA-friendly [n][128] ----------------
__global__ __launch_bounds__(128) void sample_back(const float* __restrict__ y,
                                                   const int* __restrict__ u_roll, const int* __restrict__ v_roll,
                                                   const float* __restrict__ mats, float* __restrict__ F) {
    int blk = blockIdx.x;
    int wb = blk / PIX;
    int p  = blk - wb * PIX;
    int i = p / WW, j = p % WW;
    int u = u_roll[wb], v = v_roll[wb];
    int i2 = i - u; if (i2 < 0) i2 += HH;
    int j2 = j - v; if (j2 < 0) j2 += WW;
    const float* Bm = mats + (size_t)wb * 12 + 6;

    float xs = (2.f * j2 + 1.f) / (float)WW - 1.f;
    float ys = (2.f * i2 + 1.f) / (float)HH - 1.f;
    float gx = Bm[0] * xs + Bm[1] * ys + Bm[2];
    float gy = Bm[3] * xs + Bm[4] * ys + Bm[5];
    float fx = (gx + 1.f) * (WW * 0.5f) - 0.5f;
    float fy = (gy + 1.f) * (HH * 0.5f) - 0.5f;
    float x0f = floorf(fx), y0f = floorf(fy);
    int x0 = (int)x0f, y0 = (int)y0f;
    float wx1 = fx - x0f, wx0 = 1.f - wx1;
    float wy1 = fy - y0f, wy0 = 1.f - wy1;
    bool v00 = (x0 >= 0 && x0 < WW && y0 >= 0 && y0 < HH);
    bool v10 = (x0 + 1 >= 0 && x0 + 1 < WW && y0 >= 0 && y0 < HH);
    bool v01 = (x0 >= 0 && x0 < WW && y0 + 1 >= 0 && y0 + 1 < HH);
    bool v11 = (x0 + 1 >= 0 && x0 + 1 < WW && y0 + 1 >= 0 && y0 + 1 < HH);

    int d = threadIdx.x;
    const float* yc = y + ((size_t)wb * DF + d) * PIX;
    float acc = 0.f;
    if (v00) acc += wx0 * wy0 * yc[y0 * WW + x0];
    if (v10) acc += wx1 * wy0 * yc[y0 * WW + x0 + 1];
    if (v01) acc += wx0 * wy1 * yc[(y0 + 1) * WW + x0];
    if (v11) acc += wx1 * wy1 * yc[(y0 + 1) * WW + x0 + 1];
    F[((size_t)wb * PIX + p) * DF + d] = acc;
}

// ---------------- WMMA strip GEMM: D[s] = Fa[16 rows] x Fb[16*s..16*s+15 rows]^T ----------------
// A frag (16x4 f32): lane l<16 holds row n0+l, K=k+{0,1}; lane l+16 holds K=k+{2,3}.
// B frag (4x16 f32): lane l<16 holds col m +l, K=k+{0,1}; lane l+16 holds K=k+{2,3}.
// Both are a float2 from row-major [row][128] features at +k+2*(lane>>4).
// The A fragment is loaded once per k and reused across MT subtiles (immediate offsets for s).
// One matrix at a time (MT*8 = 40 accumulator VGPRs) so the allocator can batch loads in flight.
__device__ __forceinline__ void gemm_strip(const float* __restrict__ Fa, const float* __restrict__ Fb,
                                           int lane, v8f* __restrict__ D) {
    const int half = lane >> 4, l16 = lane & 15;
    const float* pa = Fa + l16 * DF + 2 * half;
    const float* pb = Fb + l16 * DF + 2 * half;
    #pragma unroll
    for (int s = 0; s < MT; s++) D[s] = (v8f){0,0,0,0,0,0,0,0};
    #pragma unroll 4
    for (int k = 0; k < DF; k += 4) {
        v2f a = *(const v2f*)(pa + k);
        #pragma unroll
        for (int s = 0; s < MT; s++) {
            v2f bf = *(const v2f*)(pb + k + s * 16 * DF);   // +s*8192B immediate offsets
            D[s] = __builtin_amdgcn_wmma_f32_16x16x4_f32(false, a, false, bf, (short)0, D[s], false, false);
        }
    }
}

// D layout: VGPR r, lane L -> row n = n0 + r + 8*(L>>4), col m = m0 + 16*s + (L&15)

// pass 1: hmax[b][n] = max_m Mh ; vmax[b][m] = max_n Mv
__global__ __launch_bounds__(256) void corr_pass1(const float* __restrict__ F,
                                                  unsigned* __restrict__ hmax, unsigned* __restrict__ vmax) {
    int b = blockIdx.z;
    int strip = blockIdx.y * 8 + (threadIdx.x >> 5);
    if (strip >= NSTRIP) return;
    int n_t = blockIdx.x;
    int lane = threadIdx.x & 31;
    int half = lane >> 4, l16 = lane & 15;
    int n0 = n_t * 16, m0 = strip * 16 * MT;
    v8f D[MT];

    // ---- Mh = fa_h x fb_h^T : row maxes ----
    gemm_strip(F + ((size_t)(0 + b) * PIX + n0) * DF,
               F + ((size_t)(4 + b) * PIX + m0) * DF, lane, D);
    #pragma unroll
    for (int r = 0; r < 8; r++) {
        float v = D[0][r];
        #pragma unroll
        for (int s = 1; s < MT; s++) v = fmaxf(v, D[s][r]);
        v = fmaxf(v, SWZ(v, 1));
        v = fmaxf(v, SWZ(v, 2));
        v = fmaxf(v, SWZ(v, 4));
        v = fmaxf(v, SWZ(v, 8));
        if (l16 == 0) atomicMax(&hmax[b * PIX + n0 + r + 8 * half], fenc(v));
    }

    // ---- Mv = fa_v x fb_v^T : col maxes ----
    gemm_strip(F + ((size_t)(8 + b) * PIX + n0) * DF,
               F + ((size_t)(12 + b) * PIX + m0) * DF, lane, D);
    #pragma unroll
    for (int s = 0; s < MT; s++) {
        float cv = D[s][0];
        #pragma unroll
        for (int r = 1; r < 8; r++) cv = fmaxf(cv, D[s][r]);
        cv = fmaxf(cv, SWZ(cv, 16));
        if (half == 0) atomicMax(&vmax[b * PIX + m0 + 16 * s + l16], fenc(cv));
    }
}

// pass 2: hsum[b][n] = sum_m exp(Mh - hmax[n]) ; vsum[b][m] = sum_n exp(Mv - vmax[m])
__global__ __launch_bounds__(256) void corr_pass2(const float* __restrict__ F,
                                                  const unsigned* __restrict__ hmax, const unsigned* __restrict__ vmax,
                                                  float* __restrict__ hsum, float* __restrict__ vsum) {
    int b = blockIdx.z;
    int strip = blockIdx.y * 8 + (threadIdx.x >> 5);
    if (strip >= NSTRIP) return;
    int n_t = blockIdx.x;
    int lane = threadIdx.x & 31;
    int half = lane >> 4, l16 = lane & 15;
    int n0 = n_t * 16, m0 = strip * 16 * MT;
    v8f D[MT];

    // ---- Mh row sums of exp ----
    gemm_strip(F + ((size_t)(0 + b) * PIX + n0) * DF,
               F + ((size_t)(4 + b) * PIX + m0) * DF, lane, D);
    #pragma unroll
    for (int r = 0; r < 8; r++) {
        float hm = fdec(hmax[b * PIX + n0 + r + 8 * half]);
        float v = 0.f;
        #pragma unroll
        for (int s = 0; s < MT; s++) v += __expf(D[s][r] - hm);
        v += SWZ(v, 1);
        v += SWZ(v, 2);
        v += SWZ(v, 4);
        v += SWZ(v, 8);
        if (l16 == 0) atomicAdd(&hsum[b * PIX + n0 + r + 8 * half], v);
    }

    // ---- Mv col sums of exp ----
    gemm_strip(F + ((size_t)(8 + b) * PIX + n0) * DF,
               F + ((size_t)(12 + b) * PIX + m0) * DF, lane, D);
    #pragma unroll
    for (int s = 0; s < MT; s++) {
        float vm = fdec(vmax[b * PIX + m0 + 16 * s + l16]);
        float cs = 0.f;
        #pragma unroll
        for (int r = 0; r < 8; r++) cs += __expf(D[s][r] - vm);
        cs += SWZ(cs, 16);
        if (half == 0) atomicAdd(&vsum[b * PIX + m0 + 16 * s + l16], cs);
    }
}

// pass 3: S[b][m] = sum_n exp(Mv - vmax[m]) * exp(Mh - hmax[n]) * hs[n]
__global__ __launch_bounds__(256) void corr_pass3(const float* __restrict__ F,
                                                  const unsigned* __restrict__ hmax, const float* __restrict__ hsum,
                                                  const unsigned* __restrict__ vmax, float* __restrict__ S) {
    int b = blockIdx.z;
    int strip = blockIdx.y * 8 + (threadIdx.x >> 5);
    if (strip >= NSTRIP) return;
    int n_t = blockIdx.x;
    int lane = threadIdx.x & 31;
    int half = lane >> 4, l16 = lane & 15;
    int n0 = n_t * 16, m0 = strip * 16 * MT;
    v8f D[MT];

    // ---- Mh: fold exp(Mh - hmax[n]) * hs[n] into per-lane factors, kept in Eh ----
    gemm_strip(F + ((size_t)(0 + b) * PIX + n0) * DF,
               F + ((size_t)(4 + b) * PIX + m0) * DF, lane, D);
    float Eh[MT][8];
    #pragma unroll
    for (int r = 0; r < 8; r++) {
        int n = b * PIX + n0 + r + 8 * half;
        float hm = fdec(hmax[n]);
        float hsv = 1.f / (hsum[n] + 1e-4f);
        #pragma unroll
        for (int s = 0; s < MT; s++) Eh[s][r] = __expf(D[s][r] - hm) * hsv;
    }

    // ---- Mv: multiply by exp(Mv - vmax[m]) and contract over n ----
    gemm_strip(F + ((size_t)(8 + b) * PIX + n0) * DF,
               F + ((size_t)(12 + b) * PIX + m0) * DF, lane, D);
    #pragma unroll
    for (int s = 0; s < MT; s++) {
        float vm = fdec(vmax[b * PIX + m0 + 16 * s + l16]);
        float t = 0.f;
        #pragma unroll
        for (int r = 0; r < 8; r++)
            t += __expf(D[s][r] - vm) * Eh[s][r];
        t += SWZ(t, 16);
        if (half == 0) atomicAdd(&S[b * PIX + m0 + 16 * s + l16], t);
    }
}

// ---------------- finalize: mean over (b,m) of log(S * vs + 1e-4) ----------------
__global__ __launch_bounds__(256) void finalize(const float* __restrict__ S, const float* __restrict__ vsum,
                                                float* __restrict__ out) {
    __shared__ float red[256];
    float acc = 0.f;
    for (int idx = threadIdx.x; idx < BD * PIX; idx += 256) {
        float vs = 1.f / (vsum[idx] + 1e-4f);
        acc += logf(S[idx] * vs + 1e-4f);
    }
    red[threadIdx.x] = acc;
    __syncthreads();
    for (int s = 128; s > 0; s >>= 1) {
        if (threadIdx.x < s) red[threadIdx.x] += red[threadIdx.x + s];
        __syncthreads();
    }
    if (threadIdx.x == 0) out[0] = red[0] / (float)(BD * PIX);
}

extern "C" void kernel_launch(void* const* d_in, const int* in_sizes, int n_in,
                              void* d_out, int out_size, void* d_ws, size_t ws_size,
                              hipStream_t stream) {
    const float* ia    = (const float*)d_in[0];
    const float* ib    = (const float*)d_in[1];
    const float* w1    = (const float*)d_in[2];
    const float* b1    = (const float*)d_in[3];
    const float* w2    = (const float*)d_in[4];
    const float* b2    = (const float*)d_in[5];
    const float* noise = (const float*)d_in[6];
    const int* u_roll  = (const int*)d_in[7];
    const int* v_roll  = (const int*)d_in[8];
    const int* swp     = (const int*)d_in[9];
    float* ws = (float*)d_ws;

    float*    mats = ws + MATS_OFF;
    float*    xw   = ws + XW_OFF;
    float*    y    = ws + Y_OFF;
    float*    F    = ws + F_OFF;
    unsigned* hmax = (unsigned*)(ws + HMAX_OFF);
    float*    hsum = ws + HSUM_OFF;
    unsigned* vmax = (unsigned*)(ws + VMAX_OFF);
    float*    vsum = ws + VSUM_OFF;
    float*    Sarr = ws + S_OFF;

    make_mats<<<1, 16, 0, stream>>>(noise, swp, mats);
    zero_accum<<<(5 * BD * PIX + 255) / 256, 256, 0, stream>>>(ws + HMAX_OFF, 5 * BD * PIX);
    sample_front<<<(16 * PIX + 255) / 256, 256, 0, stream>>>(ia, ib, u_roll, v_roll, mats, xw);
    conv_fused<<<16 * PIX, HID, 0, stream>>>(xw, w1, b1, w2, b2, y);
    sample_back<<<16 * PIX, DF, 0, stream>>>(y, u_roll, v_roll, mats, F);

    dim3 grid(NT, (NSTRIP + 7) / 8, BD);
    corr_pass1<<<grid, 256, 0, stream>>>(F, hmax, vmax);
    corr_pass2<<<grid, 256, 0, stream>>>(F, hmax, vmax, hsum, vsum);
    corr_pass3<<<grid, 256, 0, stream>>>(F, hmax, hsum, vmax, Sarr);
    finalize<<<1, 256, 0, stream>>>(Sarr, vsum, (float*)d_out);
}